// SkipGramNS_10857677325092
// MI455X (gfx1250) — compile-verified
//
#include <hip/hip_runtime.h>
#include <math.h>

// Skip-gram negative-sampling loss for MI455X (gfx1250).
//
// v3: fixed async-builtin pointer types. Embedding-row gathers staged into
// LDS with coalesced GLOBAL_LOAD_ASYNC_TO_LDS_B128 (1 instr = 1 full 512B
// row, ASYNCcnt-tracked, double-buffered), eliminating the 8x L2 request
// fragmentation of per-lane 8B operand gathers. WMMA operands are then read
// from LDS with bank-conflict-free ds_load_b64 (row stride 132 floats ->
// banks 4p+{0..3}, all 64 once). fp32 WMMA chains (V_WMMA_F32_16X16X4_F32)
// compute 16x16 pairwise dot tiles; the diagonal gives per-element scores.
// Deterministic two-pass reduction.

#define VOCAB 100000
#define DIM   128
#define BATCH 262144
#define KNEG  10

#define WAVES_PER_BLOCK 2
#define THREADS (WAVES_PER_BLOCK * 32)
#define TILES   (BATCH / 16)                    // 16384 tiles of 16 rows
#define BLOCKS  (TILES / WAVES_PER_BLOCK)       // 8192 workgroups

#define ROWF   132                 // padded LDS row stride in floats (528 B)
#define TILEF  (16 * ROWF)         // floats per staged 16-row tile

typedef __attribute__((ext_vector_type(2))) float v2f;
typedef __attribute__((ext_vector_type(8))) float v8f;

// Pointer element type expected by the async-LDS builtin (per hipcc diag):
// 'int __attribute__((vector_size(16)))' in AS(1)/AS(3).
typedef int v4i __attribute__((vector_size(16)));
typedef __attribute__((address_space(1))) v4i* gv4i_p;
typedef __attribute__((address_space(3))) v4i* lv4i_p;

#define HAS_ASYNC_LDS __has_builtin(__builtin_amdgcn_global_load_async_to_lds_b128)

// Copy 16 contiguous bytes of a global row into LDS (per-lane slice of a
// fully coalesced 512B row fetch).
__device__ __forceinline__ void async_row_chunk(float* lds_dst, const float* g_src)
{
#if HAS_ASYNC_LDS
    __builtin_amdgcn_global_load_async_to_lds_b128(
        (gv4i_p)(uintptr_t)g_src,
        (lv4i_p)(uintptr_t)lds_dst,
        0, 0);
#else
    *(float4*)lds_dst = *(const float4*)g_src;   // b128 load + ds_store_b128
#endif
}

template <int N>
__device__ __forceinline__ void wait_async()
{
#if HAS_ASYNC_LDS
# if __has_builtin(__builtin_amdgcn_s_wait_asynccnt)
    __builtin_amdgcn_s_wait_asynccnt((unsigned short)N);
# else
    asm volatile("s_wait_asynccnt %0" :: "i"(N) : "memory");
# endif
#endif
}

__device__ __forceinline__ void wait_ds0()
{
#if __has_builtin(__builtin_amdgcn_s_wait_dscnt)
    __builtin_amdgcn_s_wait_dscnt(0);
#else
    asm volatile("s_wait_dscnt 0x0" ::: "memory");
#endif
}

// Stage 16 embedding rows (indices held one-per-lane in idx_lane, lanes 0..15)
// into an LDS tile: 16 coalesced row copies.
__device__ __forceinline__ void stage16(float* __restrict__ buf,
                                        const float* __restrict__ W,
                                        int idx_lane, int lane)
{
    #pragma unroll
    for (int i = 0; i < 16; ++i) {
        const int ridx = __shfl(idx_lane, i, 32);
        async_row_chunk(buf + i * ROWF + lane * 4,
                        W + (size_t)ridx * DIM + lane * 4);
    }
}

__global__ __launch_bounds__(THREADS) void sgns_tile_kernel(
    const int*   __restrict__ target,
    const int*   __restrict__ context,
    const int*   __restrict__ negatives,
    const float* __restrict__ Wt,
    const float* __restrict__ Wc,
    float*       __restrict__ partial)
{
    __shared__ float stage[WAVES_PER_BLOCK][2][TILEF];
    __shared__ float red[WAVES_PER_BLOCK];

    const int lane = threadIdx.x & 31;
    const int wave = threadIdx.x >> 5;
    const int tile = blockIdx.x * WAVES_PER_BLOCK + wave;

    // WMMA f32 16x16x4 operand layout (same addressing for A and B):
    // lane holds row (lane&15), float2 at column 4*kc + 2*(lane>>4).
    const int r    = lane & 15;
    const int half = lane >> 4;
    const int b    = tile * 16 + r;

    float* buf0 = &stage[wave][0][0];
    float* buf1 = &stage[wave][1][0];

    // ---- stage the 16 target rows into LDS (coalesced async copies) ----
    stage16(buf0, Wt, target[b], lane);
    wait_async<0>();

    // ---- pull T tile into registers (WMMA A layout), reused for all 11 j ----
    const int roff = r * ROWF + 2 * half;
    v2f tA[32];
    #pragma unroll
    for (int kc = 0; kc < 32; ++kc)
        tA[kc] = *(const v2f*)(buf0 + roff + kc * 4);
    wait_ds0();                        // tA reads done -> buf0 reusable

    // ---- prime the pipeline: stage j=0 (context rows) into buf0 ----
    stage16(buf0, Wc, context[b], lane);

    const bool active = (lane < 8) || (lane >= 24);
    const int  idx8   = lane & 7;
    float loss = 0.0f;

    #pragma unroll 1
    for (int j = 0; j <= KNEG; ++j) {
        if (j < KNEG) {
            // prefetch j+1's rows into the other buffer
            const int nidx = negatives[b * KNEG + j];
            float* nxt = (j & 1) ? buf0 : buf1;     // buf[(j+1)&1]
            wait_ds0();                // prior reads of nxt are complete
            stage16(nxt, Wc, nidx, lane);
            wait_async<16>();          // in-order: current buffer resident
        } else {
            wait_async<0>();           // last buffer resident
        }

        const float* cur = (j & 1) ? buf1 : buf0;   // buf[j&1]

        // 32 chained fp32 WMMAs over K=128: acc = T_tile x M_j^T (16x16 f32).
        v8f acc = {0.f, 0.f, 0.f, 0.f, 0.f, 0.f, 0.f, 0.f};
        #pragma unroll
        for (int kc = 0; kc < 32; ++kc) {
            const v2f bb = *(const v2f*)(cur + roff + kc * 4);  // ds_load_b64
            acc = __builtin_amdgcn_wmma_f32_16x16x4_f32(
                false, tA[kc], false, bb, (short)0, acc, false, false);
        }

        // Diagonal of D: lanes 0..7 -> acc[lane], lanes 24..31 -> acc[lane-24];
        // each such lane's r equals the diagonal index (its own batch row).
        float s = acc[0];
        #pragma unroll
        for (int i = 1; i < 8; ++i) s = (idx8 == i) ? acc[i] : s;

        const float x  = (j == 0) ? s : -s;
        // stable log_sigmoid(x) = min(x,0) - log1p(exp(-|x|))
        const float ls = fminf(x, 0.0f) - log1pf(expf(-fabsf(x)));
        loss += active ? ls : 0.0f;
    }

    // wave32 butterfly reduction (inactive lanes contributed 0)
    #pragma unroll
    for (int m = 16; m >= 1; m >>= 1)
        loss += __shfl_xor(loss, m, 32);

    if (lane == 0) red[wave] = loss;
    __syncthreads();
    if (threadIdx.x == 0) {
        float s = 0.f;
        #pragma unroll
        for (int w = 0; w < WAVES_PER_BLOCK; ++w) s += red[w];
        partial[blockIdx.x] = s;
    }
}

__global__ __launch_bounds__(256) void sgns_reduce_kernel(
    const float* __restrict__ partial, float* __restrict__ out)
{
    __shared__ float red[256];
    float s = 0.f;
    for (int i = threadIdx.x; i < BLOCKS; i += 256) s += partial[i];
    red[threadIdx.x] = s;
    __syncthreads();
    for (int stride = 128; stride >= 1; stride >>= 1) {
        if ((int)threadIdx.x < stride) red[threadIdx.x] += red[threadIdx.x + stride];
        __syncthreads();
    }
    if (threadIdx.x == 0) out[0] = -red[0] / (float)BATCH;
}

extern "C" void kernel_launch(void* const* d_in, const int* in_sizes, int n_in,
                              void* d_out, int out_size, void* d_ws, size_t ws_size,
                              hipStream_t stream)
{
    const int*   target    = (const int*)  d_in[0];
    const int*   context   = (const int*)  d_in[1];
    const int*   negatives = (const int*)  d_in[2];
    const float* Wt        = (const float*)d_in[3];
    const float* Wc        = (const float*)d_in[4];

    float* partial = (float*)d_ws;   // BLOCKS floats (32 KB), rewritten each call

    sgns_tile_kernel<<<BLOCKS, THREADS, 0, stream>>>(target, context, negatives,
                                                     Wt, Wc, partial);
    sgns_reduce_kernel<<<1, 256, 0, stream>>>(partial, (float*)d_out);
}